// AttentionModel_50440095924467
// MI455X (gfx1250) — compile-verified
//
#include <hip/hip_runtime.h>
#include <math.h>

#define FDIM 128
#define HNUM 8
#define ALPHA_LRELU 0.2f
#define EPS_NORM 1e-12f

typedef __attribute__((ext_vector_type(2))) float v2f;
typedef __attribute__((ext_vector_type(8))) float v8f;

// ---------------- init: m = -inf, ssum = 0 ----------------
__global__ void k_init(float* __restrict__ m, float* __restrict__ s, int n) {
    int i = blockIdx.x * blockDim.x + threadIdx.x;
    if (i < n) {
        m[i] = -__builtin_inff();
        s[i] = 0.0f;
    }
}

// ---------------- projection via V_WMMA_F32_16X16X4_F32 ----------------
// sp[n][j], j in [0,16): j<8 -> s_src[n][j] = dot(x[n], aa[j][0:128])
//                        j>=8 -> s_dst[n][j-8] = dot(x[n], aa[j-8][128:256])
// One wave per 16-row node tile. A = 16x128 (M x K), B = 128x16 (K x N), 32 K-steps of 4.
__global__ void k_proj(const float* __restrict__ x, const float* __restrict__ aa,
                       float* __restrict__ sp, int nNodes) {
    int tile = (blockIdx.x * blockDim.x + threadIdx.x) >> 5;   // global wave id
    if (tile * 16 >= nNodes) return;                           // uniform per-wave exit
    int lane = threadIdx.x & 31;
    int half = lane >> 4;     // 0: K offsets {0,1}; 1: K offsets {2,3}
    int sub  = lane & 15;     // A row / B col within tile

    int arow = tile * 16 + sub;
    if (arow >= nNodes) arow = nNodes - 1;                     // clamp (keeps EXEC full)
    const float* xr = x + (size_t)arow * FDIM;
    // B column j = sub: k-contiguous slice of aa
    const float* bc = (sub < HNUM) ? (aa + (size_t)sub * (2 * FDIM))
                                   : (aa + (size_t)(sub - HNUM) * (2 * FDIM) + FDIM);

    v8f c = {};
#pragma unroll
    for (int kk = 0; kk < FDIM / 4; ++kk) {
        int k0 = kk * 4 + half * 2;
        v2f a, b;
        a.x = xr[k0]; a.y = xr[k0 + 1];
        b.x = bc[k0]; b.y = bc[k0 + 1];
        c = __builtin_amdgcn_wmma_f32_16x16x4_f32(
                /*neg_a=*/false, a, /*neg_b=*/false, b,
                /*c_mod=*/(short)0, c, /*reuse_a=*/false, /*reuse_b=*/false);
    }

    // D layout: VGPR v -> row v (lanes 0-15) / row v+8 (lanes 16-31), col = sub
#pragma unroll
    for (int v = 0; v < 8; ++v) {
        int orow = tile * 16 + v + half * 8;
        if (orow < nNodes) sp[(size_t)orow * 16 + sub] = c[v];
    }
}

// ---------------- pass 1: segment max over rows ----------------
__global__ void k_edge_max(const int* __restrict__ row, const int* __restrict__ col,
                           const float* __restrict__ sp, float* __restrict__ m, int E) {
    int i = blockIdx.x * blockDim.x + threadIdx.x;
    if (i >= E) return;
    int r = row[i], c = col[i];
    const float* ps = sp + (size_t)r * 16;       // s_src[r][0:8]
    const float* pd = sp + (size_t)c * 16 + 8;   // s_dst[c][0:8]
    float* pm = m + (size_t)r * HNUM;
#pragma unroll
    for (int h = 0; h < HNUM; ++h) {
        float e = ps[h] + pd[h];
        e = e > 0.0f ? e : ALPHA_LRELU * e;
        __hip_atomic_fetch_max(pm + h, e, __ATOMIC_RELAXED, __HIP_MEMORY_SCOPE_AGENT);
    }
}

// ---------------- pass 2: segment sum of exp(e - m[row]) ----------------
__global__ void k_edge_sum(const int* __restrict__ row, const int* __restrict__ col,
                           const float* __restrict__ sp, const float* __restrict__ m,
                           float* __restrict__ ssum, int E) {
    int i = blockIdx.x * blockDim.x + threadIdx.x;
    if (i >= E) return;
    int r = row[i], c = col[i];
    const float* ps = sp + (size_t)r * 16;
    const float* pd = sp + (size_t)c * 16 + 8;
    const float* pm = m + (size_t)r * HNUM;
    float* pss = ssum + (size_t)r * HNUM;
#pragma unroll
    for (int h = 0; h < HNUM; ++h) {
        float e = ps[h] + pd[h];
        e = e > 0.0f ? e : ALPHA_LRELU * e;
        float ex = __expf(e - pm[h]);
        atomicAdd(pss + h, ex);
    }
}

// ---------------- pass 3: normalize and write a.T (H x E) ----------------
__global__ void k_edge_out(const int* __restrict__ row, const int* __restrict__ col,
                           const float* __restrict__ sp, const float* __restrict__ m,
                           const float* __restrict__ ssum, float* __restrict__ out, int E) {
    int i = blockIdx.x * blockDim.x + threadIdx.x;
    if (i >= E) return;
    int r = row[i], c = col[i];
    const float* ps = sp + (size_t)r * 16;
    const float* pd = sp + (size_t)c * 16 + 8;
    const float* pm = m + (size_t)r * HNUM;
    const float* pss = ssum + (size_t)r * HNUM;
#pragma unroll
    for (int h = 0; h < HNUM; ++h) {
        float e = ps[h] + pd[h];
        e = e > 0.0f ? e : ALPHA_LRELU * e;
        float ex = __expf(e - pm[h]);
        out[(size_t)h * E + i] = ex / (pss[h] + EPS_NORM);
    }
}

extern "C" void kernel_launch(void* const* d_in, const int* in_sizes, int n_in,
                              void* d_out, int out_size, void* d_ws, size_t ws_size,
                              hipStream_t stream) {
    const float* x   = (const float*)d_in[0];
    const int*   row = (const int*)d_in[1];
    const int*   col = (const int*)d_in[2];
    const float* aa  = (const float*)d_in[3];
    float* out = (float*)d_out;

    int N = in_sizes[0] / FDIM;
    int E = in_sizes[1];

    // workspace: sp (N*16 f32) | m (N*8 f32) | ssum (N*8 f32)  => 32*N floats
    float* sp = (float*)d_ws;
    float* m  = sp + (size_t)N * 16;
    float* ss = m  + (size_t)N * HNUM;

    int nm = N * HNUM;
    k_init<<<(nm + 255) / 256, 256, 0, stream>>>(m, ss, nm);

    int tiles = (N + 15) / 16;
    int nblocks = (tiles + 7) / 8;   // 8 waves (256 threads) per block, 1 tile per wave
    k_proj<<<nblocks, 256, 0, stream>>>(x, aa, sp, N);

    int eb = (E + 255) / 256;
    k_edge_max<<<eb, 256, 0, stream>>>(row, col, sp, m, E);
    k_edge_sum<<<eb, 256, 0, stream>>>(row, col, sp, m, ss, E);
    k_edge_out<<<eb, 256, 0, stream>>>(row, col, sp, m, ss, out, E);
}